// ChamferLoss_63290638074021
// MI455X (gfx1250) — compile-verified
//
#include <hip/hip_runtime.h>
#include <hip/hip_bf16.h>
#include <math.h>

typedef __attribute__((ext_vector_type(2))) float v2f;
typedef __attribute__((ext_vector_type(8))) float v8f;

#define NPTS    4096
#define NTILES  256      // NPTS / 16
#define BATCHES 16

// Row-min kernel: out[b*NPTS + i] = min_j ||X[b,i,:] - Y[b,j,:]||^2
// Uses V_WMMA_F32_16X16X4_F32 with K padded 3->4:
//   A[i] = [-2x0, -2x1, -2x2, xx_i],  B[j] = [y0, y1, y2, 1]
//   D = A*B = xx_i - 2 x.y ;  P = D + yy_j  (yy_j added post-WMMA per lane)
// Grid: (16 batches, 32 strip groups). Block: 256 threads = 8 waves,
// each wave owns 16 rows of X and scans all 256 column tiles.
__global__ __launch_bounds__(256)
void chamfer_rowmin_kernel(const float* __restrict__ X,
                           const float* __restrict__ Y,
                           float* __restrict__ outmin)
{
    // B tiles pre-swizzled into WMMA lane layout (64 KB) + yy (16 KB) = 80 KB
    __shared__ v2f   Bl[NTILES][32];
    __shared__ float yyl[NPTS];

    const int b = blockIdx.x;
    const float* __restrict__ Xb = X + (size_t)b * NPTS * 3;
    const float* __restrict__ Yb = Y + (size_t)b * NPTS * 3;

    const int t = threadIdx.x;

    // ---- precompute yy_j = ||y_j||^2 ----
    for (int j = t; j < NPTS; j += 256) {
        float y0 = Yb[j * 3 + 0];
        float y1 = Yb[j * 3 + 1];
        float y2 = Yb[j * 3 + 2];
        yyl[j] = y0 * y0 + y1 * y1 + y2 * y2;
    }
    // ---- precompute B tiles in WMMA B-matrix lane layout ----
    // B is 4x16: VGPR0 holds rows K=0 (lanes 0-15) / K=2 (lanes 16-31),
    //            VGPR1 holds rows K=1 (lanes 0-15) / K=3 (lanes 16-31).
    for (int e = t; e < NTILES * 32; e += 256) {
        int tile = e >> 5;
        int lane = e & 31;
        int col  = (tile << 4) | (lane & 15);
        float y0 = Yb[col * 3 + 0];
        float y1 = Yb[col * 3 + 1];
        float y2 = Yb[col * 3 + 2];
        v2f v;
        if (lane < 16) { v.x = y0; v.y = y1; }      // K=0, K=1
        else           { v.x = y2; v.y = 1.0f; }    // K=2, K=3 (the "1" pad)
        Bl[tile][lane] = v;
    }
    __syncthreads();

    const int wave    = t >> 5;
    const int lane    = t & 31;
    const int m       = lane & 15;
    const int rowbase = (blockIdx.y * 8 + wave) * 16;

    // ---- build A (16x4) for this wave's 16 rows; no shuffles needed ----
    float x0 = Xb[(rowbase + m) * 3 + 0];
    float x1 = Xb[(rowbase + m) * 3 + 1];
    float x2 = Xb[(rowbase + m) * 3 + 2];
    float xx = x0 * x0 + x1 * x1 + x2 * x2;
    v2f a;
    if (lane < 16) { a.x = -2.0f * x0; a.y = -2.0f * x1; }  // K=0, K=1
    else           { a.x = -2.0f * x2; a.y = xx; }          // K=2, K=3

    v8f rmin;
    #pragma unroll
    for (int r = 0; r < 8; ++r) rmin[r] = __builtin_inff();

    // ---- main loop: one WMMA per 16x16 tile of the distance matrix ----
    #pragma unroll 4
    for (int tile = 0; tile < NTILES; ++tile) {
        v2f   bv = Bl[tile][lane];
        float yv = yyl[(tile << 4) | m];
        v8f c = {};
        v8f d = __builtin_amdgcn_wmma_f32_16x16x4_f32(
            /*neg_a=*/false, a, /*neg_b=*/false, bv,
            /*c_mod=*/(short)0, c, /*reuse_a=*/false, /*reuse_b=*/false);
        #pragma unroll
        for (int r = 0; r < 8; ++r)
            rmin[r] = fminf(rmin[r], d[r] + yv);
    }

    // ---- reduce min across the 16 lanes of each half-wave ----
    // (xor masks < 16 keep lanes 0-15 / 16-31 separate, matching C/D layout)
    #pragma unroll
    for (int r = 0; r < 8; ++r) {
        float v = rmin[r];
        v = fminf(v, __shfl_xor(v, 1, 32));
        v = fminf(v, __shfl_xor(v, 2, 32));
        v = fminf(v, __shfl_xor(v, 4, 32));
        v = fminf(v, __shfl_xor(v, 8, 32));
        rmin[r] = v;
    }
    // lane 0 holds rows rowbase+0..7 (VGPR r -> row r),
    // lane 16 holds rows rowbase+8..15 (VGPR r -> row r+8)
    if (m == 0) {
        int rowoff = (lane >> 4) * 8;
        #pragma unroll
        for (int r = 0; r < 8; ++r)
            outmin[(size_t)b * NPTS + rowbase + rowoff + r] = rmin[r];
    }
}

// Final reduction: mean(minsA) + mean(minsB) over 16*4096 elements each.
__global__ __launch_bounds__(256)
void chamfer_reduce_kernel(const float* __restrict__ mins,
                           float* __restrict__ out)
{
    __shared__ float sdata[256];
    float s = 0.0f;
    for (int i = threadIdx.x; i < 2 * BATCHES * NPTS; i += 256)
        s += mins[i];
    sdata[threadIdx.x] = s;
    __syncthreads();
    for (int off = 128; off > 0; off >>= 1) {
        if (threadIdx.x < off) sdata[threadIdx.x] += sdata[threadIdx.x + off];
        __syncthreads();
    }
    if (threadIdx.x == 0)
        out[0] = sdata[0] * (1.0f / (float)(BATCHES * NPTS));
}

extern "C" void kernel_launch(void* const* d_in, const int* in_sizes, int n_in,
                              void* d_out, int out_size, void* d_ws, size_t ws_size,
                              hipStream_t stream) {
    const float* preds = (const float*)d_in[0];  // [16,4096,3] f32
    const float* gts   = (const float*)d_in[1];  // [16,4096,3] f32
    float* ws = (float*)d_ws;

    float* min_gt_to_pred = ws;                       // [16*4096]
    float* min_pred_to_gt = ws + BATCHES * NPTS;      // [16*4096]

    dim3 grid(BATCHES, NPTS / (16 * 8));  // (16, 32)
    // mins over preds for each gt point: X = gts, Y = preds
    chamfer_rowmin_kernel<<<grid, 256, 0, stream>>>(gts, preds, min_gt_to_pred);
    // mins over gts for each pred point: X = preds, Y = gts
    chamfer_rowmin_kernel<<<grid, 256, 0, stream>>>(preds, gts, min_pred_to_gt);

    chamfer_reduce_kernel<<<1, 256, 0, stream>>>(ws, (float*)d_out);
}